// ClusterLoss_19104014532699
// MI455X (gfx1250) — compile-verified
//
#include <hip/hip_runtime.h>

typedef __attribute__((ext_vector_type(2))) float v2f;
typedef __attribute__((ext_vector_type(8))) float v8f;

#define TILE_M 128            // rows per block
#define KC     128            // K chunk staged in LDS
#define LDS_STRIDE (KC + 1)   // +1 float pad -> odd word stride, bank-conflict free

// ---------------------------------------------------------------------------
// Kernel 1: sq[i] = sum_k X[i,k]^2   (one wave per row)
// ---------------------------------------------------------------------------
__global__ __launch_bounds__(256)
void cl_sqnorm_kernel(const float* __restrict__ X, float* __restrict__ sq,
                      int n, int k) {
    const int wave = threadIdx.x >> 5;
    const int lane = threadIdx.x & 31;
    const int row  = blockIdx.x * 8 + wave;
    if (row >= n) return;
    const float4* xr = (const float4*)(X + (size_t)row * k);
    const int nq = k >> 2;
    float s = 0.0f;
    for (int q = lane; q < nq; q += 32) {
        float4 v = xr[q];
        s += v.x * v.x + v.y * v.y + v.z * v.z + v.w * v.w;
    }
    #pragma unroll
    for (int off = 16; off > 0; off >>= 1) s += __shfl_xor(s, off, 32);
    if (lane == 0) sq[row] = s;
}

// ---------------------------------------------------------------------------
// Kernel 2: fused Gram(WMMA f32 16x16x4) + masked far/near + loss accumulate
// Block: 256 threads = 8 waves; block covers 128 rows, loops all column tiles.
// ---------------------------------------------------------------------------
__global__ __launch_bounds__(256)
void cl_loss_kernel(const float* __restrict__ X, const int* __restrict__ tgt,
                    const float* __restrict__ sq, float* __restrict__ out,
                    int n, int k, float invn) {
    __shared__ float ldsA[TILE_M * LDS_STRIDE];
    __shared__ float ldsB[16 * LDS_STRIDE];

    const int tid    = threadIdx.x;
    const int wave   = tid >> 5;
    const int lane   = tid & 31;
    const int lane16 = lane & 15;
    const int half   = lane >> 4;          // 0: rows r, 1: rows r+8
    const int rb     = blockIdx.x * TILE_M;
    const int rowbase = rb + wave * 16;

    // Per-lane row-side stats for the 8 accumulator rows this lane sees.
    float sqi[8]; int ti[8]; int irow[8];
    #pragma unroll
    for (int r = 0; r < 8; ++r) {
        irow[r] = rowbase + r + half * 8;
        sqi[r]  = sq[irow[r]];
        ti[r]   = tgt[irow[r]];
    }
    float farv[8], nearv[8];
    #pragma unroll
    for (int r = 0; r < 8; ++r) { farv[r] = -__builtin_inff(); nearv[r] = __builtin_inff(); }

    const int nTilesJ = n >> 4;
    const int koff = half * 2;             // WMMA f32 A/B layout: hi-half holds K+2,K+3

    for (int jt = 0; jt < nTilesJ; ++jt) {
        v8f c = {};                        // 16x16 f32 accumulator (8 VGPRs)

        for (int kc = 0; kc < k; kc += KC) {
            // ---- cooperative stage: A (128 x KC) and B (16 x KC) into LDS ----
            const int A4 = TILE_M * (KC / 4);
            const int B4 = 16 * (KC / 4);
            for (int f = tid; f < A4; f += 256) {
                int row = f >> 5;          // KC/4 == 32 float4 per row
                int c4  = f & 31;
                float4 v = *(const float4*)(X + (size_t)(rb + row) * k + kc + c4 * 4);
                float* dst = &ldsA[row * LDS_STRIDE + c4 * 4];
                dst[0] = v.x; dst[1] = v.y; dst[2] = v.z; dst[3] = v.w;
            }
            for (int f = tid; f < B4; f += 256) {
                int row = f >> 5;
                int c4  = f & 31;
                float4 v = *(const float4*)(X + (size_t)(jt * 16 + row) * k + kc + c4 * 4);
                float* dst = &ldsB[row * LDS_STRIDE + c4 * 4];
                dst[0] = v.x; dst[1] = v.y; dst[2] = v.z; dst[3] = v.w;
            }
            // speculative prefetch of next A chunk (global_prefetch_b8)
            if (kc + KC < k) {
                __builtin_prefetch(X + (size_t)(rb + (tid >> 1)) * k + (kc + KC) + (tid & 1) * 64, 0, 3);
            }
            __syncthreads();

            // ---- WMMA over this K chunk ----
            const float* aRow = &ldsA[(wave * 16 + lane16) * LDS_STRIDE];
            const float* bRow = &ldsB[lane16 * LDS_STRIDE];
            #pragma unroll 8
            for (int kk = 0; kk < KC; kk += 4) {
                v2f a, b;
                a.x = aRow[kk + koff]; a.y = aRow[kk + koff + 1];
                b.x = bRow[kk + koff]; b.y = bRow[kk + koff + 1];
                c = __builtin_amdgcn_wmma_f32_16x16x4_f32(
                        /*neg_a=*/false, a, /*neg_b=*/false, b,
                        /*c_mod=*/(short)0, c, /*reuse_a=*/false, /*reuse_b=*/false);
            }
            __syncthreads();
        }

        // ---- epilogue: dot -> distance, masked far/near update ----
        const int jcol = jt * 16 + lane16;
        const float sqj = sq[jcol];
        const int   tj  = tgt[jcol];
        #pragma unroll
        for (int r = 0; r < 8; ++r) {
            float d2 = sqi[r] + sqj - 2.0f * c[r];
            float d  = __builtin_sqrtf(fmaxf(d2, 1e-12f));
            if (ti[r] == tj) {
                farv[r] = fmaxf(farv[r], d);
                if (jcol != irow[r]) nearv[r] = fminf(nearv[r], d);
            }
        }
    }

    // ---- reduce far/near across the 16 lanes of each half (butterfly) ----
    #pragma unroll
    for (int r = 0; r < 8; ++r) {
        #pragma unroll
        for (int off = 8; off > 0; off >>= 1) {
            farv[r]  = fmaxf(farv[r],  __shfl_xor(farv[r],  off, 32));
            nearv[r] = fminf(nearv[r], __shfl_xor(nearv[r], off, 32));
        }
    }
    float part = 0.0f;
    #pragma unroll
    for (int r = 0; r < 8; ++r) part += fmaxf(farv[r] - nearv[r], 0.0f);
    // low half holds rows r..r+7, high half rows r+8..r+15 -> combine halves
    float tot = part + __shfl_xor(part, 16, 32);
    if (lane == 0) atomicAdd(out, tot * invn);
}

// ---------------------------------------------------------------------------
extern "C" void kernel_launch(void* const* d_in, const int* in_sizes, int n_in,
                              void* d_out, int out_size, void* d_ws, size_t ws_size,
                              hipStream_t stream) {
    const float* X   = (const float*)d_in[0];
    const int*   tgt = (const int*)d_in[1];
    const int n = in_sizes[1];             // 4096
    const int k = in_sizes[0] / n;         // 2048
    float* sq  = (float*)d_ws;             // n floats of scratch
    float* out = (float*)d_out;

    hipMemsetAsync(out, 0, sizeof(float), stream);
    cl_sqnorm_kernel<<<dim3(n / 8), dim3(256), 0, stream>>>(X, sq, n, k);
    cl_loss_kernel<<<dim3(n / TILE_M), dim3(256), 0, stream>>>(X, tgt, sq, out, n, k, 1.0f / (float)n);
}